// P_GCN_14078902796912
// MI455X (gfx1250) — compile-verified
//
#include <hip/hip_runtime.h>
#include <hip/hip_bf16.h>
#include <math.h>

typedef __attribute__((ext_vector_type(2))) float v2f;
typedef __attribute__((ext_vector_type(8))) float v8f;
typedef __attribute__((ext_vector_type(4))) int   v4i;

#define AS1 __attribute__((address_space(1)))
#define AS3 __attribute__((address_space(3)))

#if defined(__has_builtin)
#if __has_builtin(__builtin_amdgcn_global_load_async_to_lds_b128) && \
    __has_builtin(__builtin_amdgcn_global_load_async_to_lds_b32) &&  \
    __has_builtin(__builtin_amdgcn_s_wait_asynccnt)
#define USE_ASYNC 1
#endif
#endif
#ifndef USE_ASYNC
#define USE_ASYNC 0
#endif

constexpr int kNU = 1024;          // N_USERS
constexpr int kNI = 2048;          // N_ITEMS
constexpr int kN  = kNU + kNI;     // 3072
constexpr int kD  = 128;
constexpr int kB  = 8192;
constexpr int kK  = 64;            // top-k per row (all layers)
constexpr float kGamma = 0.2f;
constexpr float kXScale = 1.0f;

// ---------------------------------------------------------------------------
// Dense fp32 GEMM  C = A @ B  (row-major, n x n) using V_WMMA_F32_16X16X4_F32.
// 128x128 tile / 256-thread block (8 waves); wave: 16 rows x 128 cols.
// Double-buffered LDS; async global->LDS prefetch (ASYNCcnt) when available.
// B tile stored TRANSPOSED in LDS so every fragment is one aligned ds_load_b64.
// ---------------------------------------------------------------------------
#define TM 128
#define TN 128
#define TK 32
#define ASTRIDE (TK + 4)   // 36 words: even (8B-aligned v2f), conflict-free
#define BSTRIDE (TK + 2)   // 34 words: even (8B-aligned v2f), conflict-free

__global__ __launch_bounds__(256) void gemm_wmma_f32(const float* __restrict__ A,
                                                     const float* __restrict__ Bm,
                                                     float* __restrict__ C, int n) {
  __shared__ __align__(16) float As[2][TM][ASTRIDE];    // As[buf][row][k]
  __shared__ __align__(16) float BsT[2][TN][BSTRIDE];   // BsT[buf][col][k]

  const int tid  = threadIdx.x;
  const int lane = tid & 31;
  const int wave = tid >> 5;       // 0..7
  const int half = lane >> 4;      // K pair select (A/B); +8 row select (C/D)
  const int l16  = lane & 15;
  const int tileM = blockIdx.y * TM;
  const int tileN = blockIdx.x * TN;
  const int rowBase = wave * 16;

  v8f acc[8] = {};

  auto fillA = [&](int buf, int k0) {
    // 128x32 tile = 1024 float4 -> 4 instructions per wave
    for (int idx = tid; idx < TM * (TK / 4); idx += 256) {
      int r  = idx >> 3;
      int c4 = idx & 7;
      const float* src = &A[(size_t)(tileM + r) * n + k0 + c4 * 4];
      float* dst = &As[buf][r][c4 * 4];
#if USE_ASYNC
      __builtin_amdgcn_global_load_async_to_lds_b128(
          (AS1 v4i*)const_cast<float*>(src), (AS3 v4i*)dst, 0, 0);
#else
      *(float4*)dst = *(const float4*)src;
#endif
    }
  };

  auto fillB = [&](int buf, int k0) {
#if USE_ASYNC
    // transpose-scatter: 4096 b32 asyncs -> 16 instructions per wave
    for (int idx = tid; idx < TK * TN; idx += 256) {
      int c = idx & 127;
      int r = idx >> 7;
      const float* src = &Bm[(size_t)(k0 + r) * n + tileN + c];
      float* dst = &BsT[buf][c][r];
      __builtin_amdgcn_global_load_async_to_lds_b32(
          (AS1 int*)const_cast<float*>(src), (AS3 int*)dst, 0, 0);
    }
#else
    // pair rows (r2, r2+1) at one column -> one conflict-free ds_store_b64
    for (int idx = tid; idx < (TK / 2) * TN; idx += 256) {
      int c  = idx & 127;
      int r2 = (idx >> 7) * 2;
      float x0 = Bm[(size_t)(k0 + r2) * n + tileN + c];
      float x1 = Bm[(size_t)(k0 + r2 + 1) * n + tileN + c];
      *(float2*)&BsT[buf][c][r2] = make_float2(x0, x1);
    }
#endif
  };

  const int nTiles = n / TK;
  fillA(0, 0);
  fillB(0, 0);

  for (int t = 0; t < nTiles; t++) {
    const int cur = t & 1;
    const bool more = (t + 1 < nTiles);
    if (more) {                       // prefetch next tile into other buffer
      fillA(cur ^ 1, (t + 1) * TK);
      fillB(cur ^ 1, (t + 1) * TK);
    }
#if USE_ASYNC
    // async loads retire in order: 20 ops per tile fill per wave.
    if (more) __builtin_amdgcn_s_wait_asynccnt(20);  // current tile resident
    else      __builtin_amdgcn_s_wait_asynccnt(0);
#endif
    __syncthreads();                  // all waves' fills visible

    const float* aRow = &As[cur][rowBase + l16][0];
#pragma unroll
    for (int kk = 0; kk < TK; kk += 4) {
      v2f a = *(const v2f*)&aRow[kk + 2 * half];
#pragma unroll
      for (int nt = 0; nt < 8; nt++) {
        v2f b = *(const v2f*)&BsT[cur][nt * 16 + l16][kk + 2 * half];
        acc[nt] = __builtin_amdgcn_wmma_f32_16x16x4_f32(
            false, a, false, b, (short)0, acc[nt], false, false);
      }
    }
    __syncthreads();                  // done reading before next refill
  }

  // C/D layout: VGPR r -> row (r + 8*half), col = l16
#pragma unroll
  for (int nt = 0; nt < 8; nt++) {
    int col = tileN + nt * 16 + l16;
#pragma unroll
    for (int r = 0; r < 8; r++) {
      int row = tileM + rowBase + r + 8 * half;
      C[(size_t)row * n + col] = acc[nt][r];
    }
  }
}

// ---------------------------------------------------------------------------
// y = A @ x  (one block per row)
// ---------------------------------------------------------------------------
__global__ __launch_bounds__(256) void matvec_k(const float* __restrict__ A,
                                                const float* __restrict__ x,
                                                float* __restrict__ y, int n) {
  __shared__ float red[256];
  int row = blockIdx.x;
  float s = 0.f;
  for (int j = threadIdx.x; j < n; j += 256) s += A[(size_t)row * n + j] * x[j];
  red[threadIdx.x] = s;
  __syncthreads();
  for (int off = 128; off > 0; off >>= 1) {
    if (threadIdx.x < off) red[threadIdx.x] += red[threadIdx.x + off];
    __syncthreads();
  }
  if (threadIdx.x == 0) y[row] = red[0];
}

// ---------------------------------------------------------------------------
// attention weights: w = [u.v, u.v1, u.v2, u.v3]; w /= sum(w); att = softmax(w)
// ---------------------------------------------------------------------------
__global__ __launch_bounds__(256) void attention_k(const float* __restrict__ u,
                                                   const float* __restrict__ v,
                                                   const float* __restrict__ v1,
                                                   const float* __restrict__ v2,
                                                   const float* __restrict__ v3,
                                                   float* __restrict__ att, int n) {
  __shared__ float red[4][256];
  int t = threadIdx.x;
  float s0 = 0.f, s1 = 0.f, s2 = 0.f, s3 = 0.f;
  for (int j = t; j < n; j += 256) {
    float uu = u[j];
    s0 += uu * v[j];
    s1 += uu * v1[j];
    s2 += uu * v2[j];
    s3 += uu * v3[j];
  }
  red[0][t] = s0; red[1][t] = s1; red[2][t] = s2; red[3][t] = s3;
  __syncthreads();
  for (int off = 128; off > 0; off >>= 1) {
    if (t < off)
      for (int q = 0; q < 4; q++) red[q][t] += red[q][t + off];
    __syncthreads();
  }
  if (t == 0) {
    float w[4] = {red[0][0], red[1][0], red[2][0], red[3][0]};
    float s = w[0] + w[1] + w[2] + w[3];
    float m = -INFINITY;
    for (int q = 0; q < 4; q++) { w[q] /= s; m = fmaxf(m, w[q]); }
    float es = 0.f, e[4];
    for (int q = 0; q < 4; q++) { e[q] = __expf(w[q] - m); es += e[q]; }
    for (int q = 0; q < 4; q++) att[q] = e[q] / es;
  }
}

// ---------------------------------------------------------------------------
// Per-row top-64 (value + index), one block per row, iterative arg-max in LDS.
// Tie break: lower index first (matches jax.lax.top_k).
// ---------------------------------------------------------------------------
__global__ __launch_bounds__(256) void topk_k(const float* __restrict__ M,
                                              float* __restrict__ outv,
                                              int* __restrict__ outi, int n) {
  __shared__ float rowv[kN];
  __shared__ float rv[256];
  __shared__ int   ri[256];
  int row = blockIdx.x;
  int t = threadIdx.x;
  for (int j = t; j < n; j += 256) rowv[j] = M[(size_t)row * n + j];
  __syncthreads();
  for (int it = 0; it < kK; it++) {
    float best = -INFINITY;
    int bi = n;
    for (int j = t; j < n; j += 256) {
      float vv = rowv[j];
      if (vv > best) { best = vv; bi = j; }
    }
    rv[t] = best; ri[t] = bi;
    __syncthreads();
    for (int off = 128; off > 0; off >>= 1) {
      if (t < off) {
        float ov = rv[t + off]; int oi = ri[t + off];
        if (ov > rv[t] || (ov == rv[t] && oi < ri[t])) { rv[t] = ov; ri[t] = oi; }
      }
      __syncthreads();
    }
    if (t == 0) {
      outv[(size_t)row * kK + it] = rv[0];
      outi[(size_t)row * kK + it] = ri[0];
      rowv[ri[0]] = -INFINITY;
    }
    __syncthreads();
  }
}

// ---------------------------------------------------------------------------
// Fused layer combine:
// light_out[i,d] = att0*emb(i,d) + sum_l att[l+1]*( sum_j topv*emb(topi,d) + GAMMA*emb0(i,d) )
// ---------------------------------------------------------------------------
__global__ __launch_bounds__(128) void fuse_k(const float* __restrict__ ue,
                                              const float* __restrict__ ie,
                                              const float* __restrict__ ue0,
                                              const float* __restrict__ ie0,
                                              const float* __restrict__ att,
                                              const float* __restrict__ topv,
                                              const int* __restrict__ topi,
                                              float* __restrict__ LO) {
  int i = blockIdx.x;
  int d = threadIdx.x;
  float e  = (i < kNU) ? ue[(size_t)i * kD + d]  : ie[(size_t)(i - kNU) * kD + d];
  float e0 = (i < kNU) ? ue0[(size_t)i * kD + d] : ie0[(size_t)(i - kNU) * kD + d];
  float a0 = att[0], a1 = att[1], a2 = att[2], a3 = att[3];
  float aw[3] = {a1, a2, a3};
  float acc = kXScale * a0 * e;
  for (int l = 0; l < 3; l++) {
    float s = kGamma * e0;
    size_t base = ((size_t)l * kN + i) * kK;
    for (int j = 0; j < kK; j++) {
      int c = topi[base + j];
      float vv = topv[base + j];
      float ec = (c < kNU) ? ue[(size_t)c * kD + d] : ie[(size_t)(c - kNU) * kD + d];
      s += vv * ec;
    }
    acc += kXScale * aw[l] * s;
  }
  LO[(size_t)i * kD + d] = acc;
}

// ---------------------------------------------------------------------------
// out[b] = dot(light_out[users[b]], light_out[N_USERS + items[b]]) — wave per b
// ---------------------------------------------------------------------------
__global__ __launch_bounds__(256) void dot_k(const int* __restrict__ users,
                                             const int* __restrict__ items,
                                             const float* __restrict__ LO,
                                             float* __restrict__ out) {
  int wid  = (blockIdx.x * blockDim.x + threadIdx.x) >> 5;
  int lane = threadIdx.x & 31;
  if (wid >= kB) return;
  const float* pu = LO + (size_t)users[wid] * kD;
  const float* pi = LO + (size_t)(kNU + items[wid]) * kD;
  float s = 0.f;
  for (int d = lane; d < kD; d += 32) s += pu[d] * pi[d];
  for (int m = 16; m > 0; m >>= 1) s += __shfl_xor(s, m, 32);
  if (lane == 0) out[wid] = s;
}

// ---------------------------------------------------------------------------
extern "C" void kernel_launch(void* const* d_in, const int* in_sizes, int n_in,
                              void* d_out, int out_size, void* d_ws, size_t ws_size,
                              hipStream_t stream) {
  const int*   users = (const int*)d_in[0];
  const int*   items = (const int*)d_in[1];
  const float* A     = (const float*)d_in[2];
  const float* uemb  = (const float*)d_in[3];
  const float* iemb  = (const float*)d_in[4];
  const float* uemb0 = (const float*)d_in[5];
  const float* iemb0 = (const float*)d_in[6];
  const float* vu    = (const float*)d_in[7];
  const float* vvec  = (const float*)d_in[8];
  float* out = (float*)d_out;

  char* ws = (char*)d_ws;
  size_t off = 0;
  auto alloc = [&](size_t bytes) -> void* {
    void* p = ws + off;
    off += (bytes + 255) & ~(size_t)255;
    return p;
  };
  float* A2   = (float*)alloc((size_t)kN * kN * sizeof(float));
  float* A3   = (float*)alloc((size_t)kN * kN * sizeof(float));
  float* v1   = (float*)alloc((size_t)kN * sizeof(float));
  float* v2   = (float*)alloc((size_t)kN * sizeof(float));
  float* v3   = (float*)alloc((size_t)kN * sizeof(float));
  float* att  = (float*)alloc(4 * sizeof(float));
  float* topv = (float*)alloc((size_t)3 * kN * kK * sizeof(float));
  int*   topi = (int*)alloc((size_t)3 * kN * kK * sizeof(int));
  float* LO   = (float*)alloc((size_t)kN * kD * sizeof(float));

  // attention scalars: v1 = A v, v2 = A v1, v3 = A v2
  matvec_k<<<kN, 256, 0, stream>>>(A, vvec, v1, kN);
  matvec_k<<<kN, 256, 0, stream>>>(A, v1, v2, kN);
  matvec_k<<<kN, 256, 0, stream>>>(A, v2, v3, kN);
  attention_k<<<1, 256, 0, stream>>>(vu, vvec, v1, v2, v3, att, kN);

  // A2 = A @ A ; A3 = A @ A2  (fp32 WMMA)
  dim3 ggrid(kN / TN, kN / TM);
  gemm_wmma_f32<<<ggrid, 256, 0, stream>>>(A, A, A2, kN);
  gemm_wmma_f32<<<ggrid, 256, 0, stream>>>(A, A2, A3, kN);

  // top-64 per row of A, A2, A3
  topk_k<<<kN, 256, 0, stream>>>(A,  topv,                       topi,                       kN);
  topk_k<<<kN, 256, 0, stream>>>(A2, topv + (size_t)kN * kK,     topi + (size_t)kN * kK,     kN);
  topk_k<<<kN, 256, 0, stream>>>(A3, topv + (size_t)2 * kN * kK, topi + (size_t)2 * kN * kK, kN);

  // combine layers -> light_out, then gather-dot
  fuse_k<<<kN, 128, 0, stream>>>(uemb, iemb, uemb0, iemb0, att, topv, topi, LO);
  dot_k<<<(kB * 32) / 256, 256, 0, stream>>>(users, items, LO, out);
}